// ScaledDotProductAttention_33887291965753
// MI455X (gfx1250) — compile-verified
//
#include <hip/hip_runtime.h>
#include <hip/hip_bf16.h>
#include <math.h>

// ---------------------------------------------------------------------------
// ScaledDotProductAttention (diagonal variant), MI455X / gfx1250, wave32.
//   B=4, S=4096, D=64
//   out  = softmax(rowdot(Q,K)/8/temp)[...,None] * V          (4  MB)
//   attn = diag_embed(softmax(...))                           (256 MB, ~zeros)
// Memory-bound: ~272 MB traffic -> ~11.7 us @ 23.3 TB/s. Zero-fill dominates.
// ---------------------------------------------------------------------------

#define BB 4
#define SS 4096
#define DD 64

typedef __attribute__((ext_vector_type(2))) float v2f;
typedef __attribute__((ext_vector_type(4))) float v4f;
typedef __attribute__((ext_vector_type(8))) float v8f;

// ---------------------------------------------------------------------------
// Kernel 1: diag[row] = <Q[row,:], K[row,:]> via V_WMMA_F32_16X16X4_F32.
// One wave32 handles 16 consecutive rows. A = Q tile (16x4), B = K^T tile
// (4x16), accumulate K=64 in 16 WMMA steps; result diagonal = row dots.
//
// A layout (ISA 7.12.2, 32-bit A 16x4): lane r (0-15) holds {K=k0, K=k0+1},
// lane r+16 holds {K=k0+2, K=k0+3} for M=r  -> contiguous float2 per lane.
// B layout (rows striped across lanes): VGPR0 = rows K=k0 (lanes 0-15) /
// K=k0+2 (lanes 16-31), VGPR1 = K=k0+1 / k0+3, with N=lane -> identical
// per-lane float2 addressing on the K matrix (B[k,n] = Ktile[n,k]).
// C/D layout: VGPR v holds (M=v, N=lane) lanes 0-15 and (M=v+8, N=lane-16)
// lanes 16-31 -> diag element v lives in lane v (rows 0-7) / lane v+24 (8-15).
// ---------------------------------------------------------------------------
__global__ __launch_bounds__(256) void sdpa_diag_wmma_kernel(
    const float* __restrict__ q, const float* __restrict__ k,
    float* __restrict__ diag) {
  const int lane = threadIdx.x & 31;
  const int wave = (int)(blockIdx.x * (blockDim.x >> 5) + (threadIdx.x >> 5));
  const long rowBase = (long)wave * 16;

  const int r = lane & 15;              // row within 16-row tile
  const int khalf = (lane >> 4) << 1;   // 0 (lanes 0-15) or 2 (lanes 16-31)
  const float* qrow = q + (rowBase + r) * DD;
  const float* krow = k + (rowBase + r) * DD;

  v8f c = {};
#pragma unroll
  for (int k0 = 0; k0 < DD; k0 += 4) {
    v2f a = *(const v2f*)(qrow + k0 + khalf);
    v2f b = *(const v2f*)(krow + k0 + khalf);
    // 8 args: (neg_a, A, neg_b, B, c_mod, C, reuse_a, reuse_b)
    c = __builtin_amdgcn_wmma_f32_16x16x4_f32(
        false, a, false, b, (short)0, c, false, false);
  }

  // Extract the 16 diagonal elements (after all WMMAs; EXEC full during WMMA).
#pragma unroll
  for (int v = 0; v < 8; ++v) {
    if (lane == v)      diag[rowBase + v]     = c[v];
    if (lane == v + 24) diag[rowBase + 8 + v] = c[v];
  }
}

// ---------------------------------------------------------------------------
// Kernel 2: per-batch softmax over S=4096. One 1024-thread block per batch,
// 4 rows/thread, LDS tree reductions for max and sum.
// ---------------------------------------------------------------------------
__global__ __launch_bounds__(1024) void sdpa_softmax_kernel(
    const float* __restrict__ diag, const float* __restrict__ temp,
    float* __restrict__ w) {
  __shared__ float red[1024];
  const int b = blockIdx.x;
  const int tid = threadIdx.x;
  const float invscale = 1.0f / (8.0f * temp[0]);  // /sqrt(64)/temp

  float x[4];
  float lmax = -INFINITY;
#pragma unroll
  for (int i = 0; i < 4; ++i) {
    x[i] = diag[b * SS + tid + i * 1024] * invscale;
    lmax = fmaxf(lmax, x[i]);
  }
  red[tid] = lmax;
  __syncthreads();
  for (int s = 512; s > 0; s >>= 1) {
    if (tid < s) red[tid] = fmaxf(red[tid], red[tid + s]);
    __syncthreads();
  }
  const float m = red[0];
  __syncthreads();

  float e[4];
  float lsum = 0.0f;
#pragma unroll
  for (int i = 0; i < 4; ++i) {
    e[i] = __expf(x[i] - m);
    lsum += e[i];
  }
  red[tid] = lsum;
  __syncthreads();
  for (int s = 512; s > 0; s >>= 1) {
    if (tid < s) red[tid] += red[tid + s];
    __syncthreads();
  }
  const float inv = 1.0f / red[0];
#pragma unroll
  for (int i = 0; i < 4; ++i) w[b * SS + tid + i * 1024] = e[i] * inv;
}

// ---------------------------------------------------------------------------
// Kernel 3: out[row, :] = w[row] * value[row, :], float4-wide. 16 v4f / row.
// ---------------------------------------------------------------------------
__global__ __launch_bounds__(256) void sdpa_out_kernel(
    const v4f* __restrict__ value, const float* __restrict__ w,
    v4f* __restrict__ out) {
  const int t = (int)(blockIdx.x * blockDim.x + threadIdx.x);  // B*S*16
  const int row = t >> 4;
  const float wv = w[row];
  v4f x = value[t];
  x *= wv;
  out[t] = x;
}

// ---------------------------------------------------------------------------
// Kernel 4: zero the 256 MB attention slab with non-temporal b128 stores
// (slab exceeds 192 MB L2 -> stream past it). Grid-stride, coalesced.
// ---------------------------------------------------------------------------
__global__ __launch_bounds__(256) void sdpa_zero_attn_kernel(
    v4f* __restrict__ attn, long n4) {
  const long idx = (long)blockIdx.x * blockDim.x + threadIdx.x;
  const long stride = (long)gridDim.x * blockDim.x;
  const v4f z = {0.0f, 0.0f, 0.0f, 0.0f};
  for (long i = idx; i < n4; i += stride)
    __builtin_nontemporal_store(z, attn + i);
}

// ---------------------------------------------------------------------------
// Kernel 5: scatter w onto the diagonal: attn[b, s, s] = w[b, s].
// ---------------------------------------------------------------------------
__global__ __launch_bounds__(256) void sdpa_diag_scatter_kernel(
    const float* __restrict__ w, float* __restrict__ attn) {
  const int t = (int)(blockIdx.x * blockDim.x + threadIdx.x);  // B*S
  const int b = t >> 12;
  const int s = t & (SS - 1);
  attn[(long)b * SS * SS + (long)s * (SS + 1)] = w[t];
}

extern "C" void kernel_launch(void* const* d_in, const int* in_sizes, int n_in,
                              void* d_out, int out_size, void* d_ws,
                              size_t ws_size, hipStream_t stream) {
  const float* q    = (const float*)d_in[0];
  const float* k    = (const float*)d_in[1];
  const float* v    = (const float*)d_in[2];
  const float* temp = (const float*)d_in[3];

  float* out  = (float*)d_out;                       // [B,S,D]
  float* attn = out + (size_t)BB * SS * DD;          // [B,S,S]

  float* diag = (float*)d_ws;                        // B*S floats
  float* w    = diag + (size_t)BB * SS;              // B*S floats

  // 1) diag dot products: B*S/16 = 1024 waves -> 128 blocks x 256 thr (8 waves)
  sdpa_diag_wmma_kernel<<<128, 256, 0, stream>>>(q, k, diag);

  // 2) softmax: one block per batch
  sdpa_softmax_kernel<<<BB, 1024, 0, stream>>>(diag, temp, w);

  // 3) out = w * value: B*S*D/4 = 262144 lanes
  sdpa_out_kernel<<<(BB * SS * DD / 4) / 256, 256, 0, stream>>>(
      (const v4f*)v, w, (v4f*)out);

  // 4) zero attention slab: 16,777,216 float4 chunks, grid-stride (16 it/thr)
  const long n4 = (long)BB * SS * SS / 4;
  sdpa_zero_attn_kernel<<<4096, 256, 0, stream>>>((v4f*)attn, n4);

  // 5) write diagonal (after zero-fill; same-stream ordering guarantees this)
  sdpa_diag_scatter_kernel<<<(BB * SS) / 256, 256, 0, stream>>>(w, attn);
}